// QTAttA_19404662243966
// MI455X (gfx1250) — compile-verified
//
#include <hip/hip_runtime.h>
#include <cmath>

// QTAttA quadtree attention for gfx1250 (MI455X), wave32.
// Level 0 GEMMs use V_WMMA_F32_16X16X4_F32 chained over d=16 (4 steps of K=4),
// matching the f32 reference exactly. Q/K/V staged in LDS via
// GLOBAL_LOAD_ASYNC_TO_LDS_B128 (ASYNCcnt path) so all WMMA operands come
// from LDS -> no scratch spills. Fine levels are gather-bound -> VALU.

typedef __attribute__((ext_vector_type(2))) float v2f;
typedef __attribute__((ext_vector_type(8))) float v8f;
typedef __attribute__((ext_vector_type(4))) int   v4i;

constexpr int NB = 4;    // batch
constexpr int NC = 128;  // channels
constexpr int NH = 8;    // heads
constexpr int ND = 16;   // head dim  (scale = 1/sqrt(16) = 0.25)

// ---- CDNA5 async global->LDS staging (ASYNCcnt path), guarded -------------
#if defined(__has_builtin)
#if __has_builtin(__builtin_amdgcn_global_load_async_to_lds_b128)
#define HAVE_ASYNC_LDS 1
#endif
#endif

typedef __attribute__((address_space(1))) v4i Gv4i;   // global int4
typedef __attribute__((address_space(3))) v4i Lv4i;   // LDS int4

__device__ __forceinline__ void lds_copy16(float* l, const float* g) {
#ifdef HAVE_ASYNC_LDS
  __builtin_amdgcn_global_load_async_to_lds_b128((Gv4i*)g, (Lv4i*)l, 0, 0);
#else
  *(float4*)l = *(const float4*)g;
#endif
}

__device__ __forceinline__ void lds_copy_wait() {
#ifdef HAVE_ASYNC_LDS
#if __has_builtin(__builtin_amdgcn_s_wait_asynccnt)
  __builtin_amdgcn_s_wait_asynccnt(0);
#else
  asm volatile("s_wait_asynccnt 0x0" ::: "memory");
#endif
#endif
}

// ---------------------------------------------------------------------------
// Level 0: dense 256x256 attention per (b,n), softmax over keys, top-16
// select+zero, residual message = A_masked * V.
// grid = NB*NH blocks, block = 128 threads (4 waves).
// static LDS: Q/K/V stage 48KB + 4 score slabs 64KB = 112KB (<320KB/WGP).
// ---------------------------------------------------------------------------
__global__ void l0_coarse(const float* __restrict__ q0, const float* __restrict__ k0,
                          const float* __restrict__ v0,
                          float* __restrict__ msg0, float* __restrict__ tsc0,
                          int* __restrict__ tix0)
{
  constexpr int S = 256;                      // 16x16 tokens
  __shared__ __align__(16) float Qs[16 * 256];      // [dc][s]
  __shared__ __align__(16) float Ks[16 * 256];      // [dc][s]
  __shared__ __align__(16) float Vs[16 * 256];      // [dc][s]
  __shared__ __align__(16) float scb[4 * 16 * 256]; // per-wave 16x256 scores

  const int bn = blockIdx.x, b = bn >> 3, n = bn & 7;
  const int wave = threadIdx.x >> 5, lane = threadIdx.x & 31;
  const int lhi = lane >> 4, llo = lane & 15;
  float* sc = scb + wave * (16 * 256);

  // channel-major token views: X[dc][s] at base + dc*S + s
  const float* qb = q0 + (size_t)(b * NC + n * ND) * S;
  const float* kb = k0 + (size_t)(b * NC + n * ND) * S;
  const float* vb = v0 + (size_t)(b * NC + n * ND) * S;
  const size_t obase = (size_t)(b * NH + n) * S;

  // ---- stage Q,K,V into LDS (3 x 16KB), async global->LDS ----
  for (int i = threadIdx.x; i < 1024; i += 128) {
    lds_copy16(Qs + 4 * i, qb + 4 * i);
    lds_copy16(Ks + 4 * i, kb + 4 * i);
    lds_copy16(Vs + 4 * i, vb + 4 * i);
  }
  lds_copy_wait();
  __syncthreads();

  for (int mt = wave; mt < 16; mt += 4) {     // 16 M-tiles, 4 waves
    const int m = mt * 16 + llo;              // query row (A frag: M = lane%16)

    // ---- scores: S[m][key] = sum_d Q[m][d]*K[key][d], chained K=4 WMMA ----
    for (int nt = 0; nt < 16; ++nt) {
      v8f acc = {};
      const int key = nt * 16 + llo;          // B frag: N = lane%16
      for (int cc = 0; cc < 4; ++cc) {
        const int kd = cc * 4 + 2 * lhi;      // K element pair per lane-half
        v2f a, bvv;
        a.x   = Qs[kd * 256 + m];
        a.y   = Qs[(kd + 1) * 256 + m];
        bvv.x = Ks[kd * 256 + key];
        bvv.y = Ks[(kd + 1) * 256 + key];
        acc = __builtin_amdgcn_wmma_f32_16x16x4_f32(false, a, false, bvv,
                                                    (short)0, acc, false, false);
      }
      // D frag: D[r + 8*lhi][llo]
      for (int r = 0; r < 8; ++r)
        sc[(r + 8 * lhi) * 256 + nt * 16 + llo] = acc[r] * 0.25f;
    }
    __syncthreads();

    // ---- softmax over 256 keys per row, then iterative top-16 + zero ----
    for (int r = 0; r < 16; ++r) {
      float* row = sc + r * 256;
      float mx = -1e30f;
      for (int i = 0; i < 8; ++i) mx = fmaxf(mx, row[lane + 32 * i]);
      for (int off = 16; off; off >>= 1) mx = fmaxf(mx, __shfl_xor(mx, off, 32));
      float ev[8], sum = 0.f;
      for (int i = 0; i < 8; ++i) { ev[i] = __expf(row[lane + 32 * i] - mx); sum += ev[i]; }
      for (int off = 16; off; off >>= 1) sum += __shfl_xor(sum, off, 32);
      const float inv = 1.f / sum;
      for (int i = 0; i < 8; ++i) { ev[i] *= inv; row[lane + 32 * i] = ev[i]; }

      const int mg = mt * 16 + r;
      for (int it = 0; it < 16; ++it) {       // top-16: record score, zero entry
        float bv = -1.f; int bi = 1 << 30;
        for (int i = 0; i < 8; ++i) {
          const int ci = lane + 32 * i;
          if (ev[i] > bv) { bv = ev[i]; bi = ci; }
        }
        for (int off = 16; off; off >>= 1) {
          float ov = __shfl_xor(bv, off, 32);
          int   oi = __shfl_xor(bi, off, 32);
          if (ov > bv || (ov == bv && oi < bi)) { bv = ov; bi = oi; }
        }
        if (lane == 0) {
          tsc0[(obase + mg) * 16 + it] = bv;
          tix0[(obase + mg) * 16 + it] = bi;
        }
        if ((bi & 31) == lane) { ev[bi >> 5] = -1.f; row[bi] = 0.f; }  // owner zeroes
      }
    }
    __syncthreads();

    // ---- msg = A_masked(16x256) * V(256x16) via chained K=4 WMMA ----
    v8f acc = {};
    for (int st = 0; st < 16; ++st)
      for (int cc = 0; cc < 4; ++cc) {
        const int ks = st * 16 + cc * 4 + 2 * lhi;   // K (= key) element pair
        v2f a, bvv;
        a.x   = sc[llo * 256 + ks];                  // A[M=llo][ks]
        a.y   = sc[llo * 256 + ks + 1];
        bvv.x = Vs[llo * 256 + ks];                  // V[ks][dc=llo]
        bvv.y = Vs[llo * 256 + ks + 1];
        acc = __builtin_amdgcn_wmma_f32_16x16x4_f32(false, a, false, bvv,
                                                    (short)0, acc, false, false);
      }
    for (int r = 0; r < 8; ++r) {
      const int mo = mt * 16 + r + 8 * lhi;
      msg0[(obase + mo) * 16 + llo] = acc[r];
    }
    __syncthreads();
  }
}

// ---------------------------------------------------------------------------
// Level 1 (fine, not final): per (b, n, quad L in 16x16): gather 16 parents x 4
// children = 64 keys; 4-child softmax * parent score; top-8 mask; aggregate.
// grid = NB*NH*256, block = 64 threads.
// ---------------------------------------------------------------------------
__global__ void l1_fine(const float* __restrict__ q1, const float* __restrict__ k1,
                        const float* __restrict__ v1,
                        const float* __restrict__ tsc0, const int* __restrict__ tix0,
                        float* __restrict__ msg1, float* __restrict__ tsc1,
                        int* __restrict__ tix1)
{
  constexpr int W = 32, S = 1024, K4 = 64;
  __shared__ float Qq[4][16];
  __shared__ float qk[4][K4];
  __shared__ float tsp[16];
  __shared__ int   ktok[K4];
  __shared__ float Vg[K4][16];
  __shared__ float topv[4][8];
  __shared__ int   topi[4][8];

  const int blk = blockIdx.x;
  const int L = blk & 255, bn = blk >> 8;
  const int b = bn >> 3, n = bn & 7;
  const int tid = threadIdx.x;
  const int Y = L >> 4, X = L & 15;

  const float* qbase = q1 + (size_t)(b * NC + n * ND) * S;
  const float* kbase = k1 + (size_t)(b * NC + n * ND) * S;
  const float* vbase = v1 + (size_t)(b * NC + n * ND) * S;
  const size_t pbase = ((size_t)(b * NH + n) * 256 + L) * 16;

  { // quad queries: member t=(t1,t2) at (2Y+t1, 2X+t2)
    const int t = tid >> 4, dc = tid & 15;
    const int y = 2 * Y + (t >> 1), x = 2 * X + (t & 1);
    Qq[t][dc] = qbase[(size_t)dc * S + y * W + x];
  }
  if (tid < 16) tsp[tid] = tsc0[pbase + tid];
  { // keys: kp = k>>2, child = k&3 = (y<<1)|x; gather V rows into LDS
    const int k = tid, kp = k >> 2, ch = k & 3;
    const int pidx = tix0[pbase + kp];          // level-0 token on 16x16
    const int p0 = (pidx >> 4) * 2, p1 = (pidx & 15) * 2;
    const int tok = (p0 + (ch >> 1)) * W + (p1 + (ch & 1));
    ktok[k] = tok;
    for (int dc = 0; dc < 16; ++dc) Vg[k][dc] = vbase[(size_t)dc * S + tok];
  }
  __syncthreads();
  { // qk[t][k] = (Q_t . K_k) / sqrt(d)
    const int k = tid, tok = ktok[k];
    float kv[16];
    for (int dc = 0; dc < 16; ++dc) kv[dc] = kbase[(size_t)dc * S + tok];
    for (int t = 0; t < 4; ++t) {
      float s = 0.f;
      for (int dc = 0; dc < 16; ++dc) s += Qq[t][dc] * kv[dc];
      qk[t][k] = s * 0.25f;
    }
  }
  __syncthreads();
  { // softmax over 4 children per parent, times parent score
    const int t = tid >> 4, kp = tid & 15;
    float a0 = qk[t][kp*4+0], a1 = qk[t][kp*4+1], a2 = qk[t][kp*4+2], a3 = qk[t][kp*4+3];
    float mx = fmaxf(fmaxf(a0, a1), fmaxf(a2, a3));
    float e0 = __expf(a0-mx), e1 = __expf(a1-mx), e2 = __expf(a2-mx), e3 = __expf(a3-mx);
    float inv = tsp[kp] / (e0 + e1 + e2 + e3);
    qk[t][kp*4+0] = e0*inv; qk[t][kp*4+1] = e1*inv;
    qk[t][kp*4+2] = e2*inv; qk[t][kp*4+3] = e3*inv;
  }
  __syncthreads();
  if (tid < 4) { // top-8 per query: record (post parent-mult), then zero
    const int t = tid;
    for (int it = 0; it < 8; ++it) {
      float bv = -1.f; int bi = 0;
      for (int k = 0; k < K4; ++k) if (qk[t][k] > bv) { bv = qk[t][k]; bi = k; }
      topv[t][it] = bv; topi[t][it] = ktok[bi];
      qk[t][bi] = 0.f;
    }
  }
  __syncthreads();
  { // residual message from masked A; quad_to_grid on the way out
    const int t = tid >> 4, dc = tid & 15;
    float s = 0.f;
    for (int k = 0; k < K4; ++k) s += qk[t][k] * Vg[k][dc];
    const int tok = (2*Y + (t>>1)) * W + (2*X + (t&1));
    msg1[((size_t)(b*NH+n)*S + tok) * 16 + dc] = s;
  }
  if (tid < 32) {
    const int t = tid >> 3, j = tid & 7;
    const int tok = (2*Y + (t>>1)) * W + (2*X + (t&1));
    tsc1[((size_t)(b*NH+n)*S + tok) * 8 + j] = topv[t][j];
    tix1[((size_t)(b*NH+n)*S + tok) * 8 + j] = topi[t][j];
  }
}

// ---------------------------------------------------------------------------
// Level 2 (fine, FINAL): 8 parents x 4 children = 32 keys; no masking.
// grid = NB*NH*1024, block = 64 threads.
// ---------------------------------------------------------------------------
__global__ void l2_fine(const float* __restrict__ q2, const float* __restrict__ k2,
                        const float* __restrict__ v2,
                        const float* __restrict__ tsc1, const int* __restrict__ tix1,
                        float* __restrict__ msg2)
{
  constexpr int W = 64, S = 4096, K4 = 32;
  __shared__ float Qq[4][16];
  __shared__ float qk[4][K4];
  __shared__ float tsp[8];
  __shared__ int   ktok[K4];
  __shared__ float Vg[K4][16];

  const int blk = blockIdx.x;
  const int L = blk & 1023, bn = blk >> 10;
  const int b = bn >> 3, n = bn & 7;
  const int tid = threadIdx.x;
  const int Y = L >> 5, X = L & 31;

  const float* qbase = q2 + (size_t)(b * NC + n * ND) * S;
  const float* kbase = k2 + (size_t)(b * NC + n * ND) * S;
  const float* vbase = v2 + (size_t)(b * NC + n * ND) * S;
  const size_t pbase = ((size_t)(b * NH + n) * 1024 + L) * 8;  // parent token == L

  {
    const int t = tid >> 4, dc = tid & 15;
    const int y = 2 * Y + (t >> 1), x = 2 * X + (t & 1);
    Qq[t][dc] = qbase[(size_t)dc * S + y * W + x];
  }
  if (tid < 8) tsp[tid] = tsc1[pbase + tid];
  if (tid < K4) {
    const int k = tid, kp = k >> 2, ch = k & 3;
    const int pidx = tix1[pbase + kp];          // level-1 token on 32x32
    const int p0 = (pidx >> 5) * 2, p1 = (pidx & 31) * 2;
    const int tok = (p0 + (ch >> 1)) * W + (p1 + (ch & 1));
    ktok[k] = tok;
    for (int dc = 0; dc < 16; ++dc) Vg[k][dc] = vbase[(size_t)dc * S + tok];
  }
  __syncthreads();
  if (tid < K4) {
    const int k = tid, tok = ktok[k];
    float kv[16];
    for (int dc = 0; dc < 16; ++dc) kv[dc] = kbase[(size_t)dc * S + tok];
    for (int t = 0; t < 4; ++t) {
      float s = 0.f;
      for (int dc = 0; dc < 16; ++dc) s += Qq[t][dc] * kv[dc];
      qk[t][k] = s * 0.25f;
    }
  }
  __syncthreads();
  if (tid < 32) { // (t, kp) pairs: 4 * 8
    const int t = tid >> 3, kp = tid & 7;
    float a0 = qk[t][kp*4+0], a1 = qk[t][kp*4+1], a2 = qk[t][kp*4+2], a3 = qk[t][kp*4+3];
    float mx = fmaxf(fmaxf(a0, a1), fmaxf(a2, a3));
    float e0 = __expf(a0-mx), e1 = __expf(a1-mx), e2 = __expf(a2-mx), e3 = __expf(a3-mx);
    float inv = tsp[kp] / (e0 + e1 + e2 + e3);
    qk[t][kp*4+0] = e0*inv; qk[t][kp*4+1] = e1*inv;
    qk[t][kp*4+2] = e2*inv; qk[t][kp*4+3] = e3*inv;
  }
  __syncthreads();
  { // final: full (unmasked) aggregation
    const int t = tid >> 4, dc = tid & 15;
    float s = 0.f;
    for (int k = 0; k < K4; ++k) s += qk[t][k] * Vg[k][dc];
    const int tok = (2*Y + (t>>1)) * W + (2*X + (t&1));
    msg2[((size_t)(b*NH+n)*S + tok) * 16 + dc] = s;
  }
}

// ---------------------------------------------------------------------------
// Fusion: out[b][n*16+dc][y][x] = msg2 + upsample2(msg1) + upsample4(msg0)
// ---------------------------------------------------------------------------
__global__ void fuse(const float* __restrict__ msg0, const float* __restrict__ msg1,
                     const float* __restrict__ msg2, float* __restrict__ out, int total)
{
  const int idx = blockIdx.x * blockDim.x + threadIdx.x;
  if (idx >= total) return;
  const int x = idx & 63, y = (idx >> 6) & 63, ch = (idx >> 12) & 127, b = idx >> 19;
  const int n = ch >> 4, dc = ch & 15;
  const size_t hb = (size_t)(b * NH + n);
  out[idx] = msg2[(hb * 4096 + (y * 64 + x)) * 16 + dc]
           + msg1[(hb * 1024 + ((y >> 1) * 32 + (x >> 1))) * 16 + dc]
           + msg0[(hb * 256  + ((y >> 2) * 16 + (x >> 2))) * 16 + dc];
}

// ---------------------------------------------------------------------------
extern "C" void kernel_launch(void* const* d_in, const int* in_sizes, int n_in,
                              void* d_out, int out_size, void* d_ws, size_t ws_size,
                              hipStream_t stream)
{
  const float* q0 = (const float*)d_in[0];
  const float* k0 = (const float*)d_in[1];
  const float* v0 = (const float*)d_in[2];
  const float* q1 = (const float*)d_in[3];
  const float* k1 = (const float*)d_in[4];
  const float* v1 = (const float*)d_in[5];
  const float* q2 = (const float*)d_in[6];
  const float* k2 = (const float*)d_in[7];
  const float* v2 = (const float*)d_in[8];
  float* out = (float*)d_out;

  // workspace partition (floats): msg0|tsc0|tix0|msg1|tsc1|tix1|msg2
  float* ws   = (float*)d_ws;
  float* msg0 = ws;                                // 4*8*256*16  = 131072
  float* tsc0 = msg0 + 131072;                     // 131072
  int*   tix0 = (int*)(tsc0 + 131072);             // 131072
  float* msg1 = (float*)(tix0 + 131072);           // 4*8*1024*16 = 524288
  float* tsc1 = msg1 + 524288;                     // 4*8*1024*8  = 262144
  int*   tix1 = (int*)(tsc1 + 262144);             // 262144
  float* msg2 = (float*)(tix1 + 262144);           // 4*8*4096*16 = 2097152

  l0_coarse<<<NB * NH, 128, 0, stream>>>(q0, k0, v0, msg0, tsc0, tix0);
  l1_fine<<<NB * NH * 256, 64, 0, stream>>>(
      q1, k1, v1, tsc0, tix0, msg1, tsc1, tix1);
  l2_fine<<<NB * NH * 1024, 64, 0, stream>>>(
      q2, k2, v2, tsc1, tix1, msg2);
  const int total = NB * NC * 64 * 64;
  fuse<<<(total + 255) / 256, 256, 0, stream>>>(msg0, msg1, msg2, out, total);
}